// MultiImage_Fuse_Sampler_89481348644941
// MI455X (gfx1250) — compile-verified
//
#include <hip/hip_runtime.h>

typedef float v2f __attribute__((ext_vector_type(2)));
typedef float v4f __attribute__((ext_vector_type(4)));
typedef float v8f __attribute__((ext_vector_type(8)));

#define BATCH 2
#define NIMG  5
#define RESO  32
#define NVOX  32768
#define CIN   128
#define CIMG  1280
#define NHD   8
#define HD    16

// pe frequency: div_i = exp(-i * ln(10000)/64)
#define PE_C  0.14391156516f

static __device__ __forceinline__ v8f wmma_f32(v2f a, v2f b, v8f c) {
  // D(16x16,f32) = A(16x4,f32) * B(4x16,f32) + C   (VOP3P, wave32)
  return __builtin_amdgcn_wmma_f32_16x16x4_f32(false, a, false, b, (short)0, c,
                                               false, false);
}

// A row-major tile (16 x K, leading dim lda), two 16-col B tiles from weight
// matrix W stored row-major as W[n][k] (B[k][n] = W[n][k]).
static __device__ __forceinline__ void gemm_loop(const float* A, int lda,
                                                 const float* W0, const float* W1,
                                                 int ldw, int K, int lane,
                                                 v8f& c0, v8f& c1) {
  const int mr = lane & 15;
  const int kg = (lane >> 4) << 1;   // 0 or 2
  const float* ap  = A  + mr * lda + kg;
  const float* b0p = W0 + mr * ldw + kg;
  const float* b1p = W1 + mr * ldw + kg;
#pragma unroll 4
  for (int k0 = 0; k0 < K; k0 += 4) {
    v2f a  = *(const v2f*)(ap  + k0);
    v2f b0 = *(const v2f*)(b0p + k0);
    v2f b1 = *(const v2f*)(b1p + k0);
    c0 = wmma_f32(a, b0, c0);
    c1 = wmma_f32(a, b1, c1);
  }
}

// ---------------- Kernel 1: k_feat / v_feat = img @ {k,v}_w.T + b + pe ------
__global__ __launch_bounds__(256) void kv_feat_kernel(
    const float* __restrict__ image_feat, const float* __restrict__ k_w,
    const float* __restrict__ k_b, const float* __restrict__ v_w,
    const float* __restrict__ v_b, float* __restrict__ k_feat,
    float* __restrict__ v_feat) {
  const int lane = threadIdx.x & 31;
  const int gw   = blockIdx.x * 8 + (threadIdx.x >> 5);   // [0,1280)
  const int mt = gw >> 3, rem = gw & 7;
  const int kv = rem >> 2, np = rem & 3, n0 = np * 32;
  const int r0 = mt * 16;
  const int b = r0 / (NIMG * 256), j = (r0 / 256) % NIMG, p0 = r0 & 255;
  const float* A = image_feat + (size_t)((b * NIMG + j) * 257 + 1 + p0) * CIMG;
  const float* W = kv ? v_w : k_w;
  const float* bias = kv ? v_b : k_b;
  float* out = (kv ? v_feat : k_feat) + (size_t)r0 * CIN;

  v8f c0 = {0,0,0,0,0,0,0,0}, c1 = c0;
  gemm_loop(A, CIMG, W + n0 * CIMG, W + (n0 + 16) * CIMG, CIMG, CIMG, lane, c0, c1);

  const int mr = lane & 15, mhi = (lane >> 4) << 3;
#pragma unroll
  for (int t = 0; t < 2; t++) {
    const v8f acc = t ? c1 : c0;
    const int col = n0 + t * 16 + mr;
    const float bb = bias[col];
    const float div = __expf(-PE_C * (float)(col >> 1));
    const bool odd = col & 1;
#pragma unroll
    for (int v = 0; v < 8; v++) {
      const int m = v + mhi;
      const float ang = (float)(p0 + m) * div;
      const float pe = odd ? cosf(ang) : sinf(ang);
      out[m * CIN + col] = acc[v] + bb + pe;
    }
  }
}

// ------- Kernel 2: fold head projection into maps: kh_map = k_feat@Wk.T+bk --
__global__ __launch_bounds__(256) void kv_head_kernel(
    const float* __restrict__ k_feat, const float* __restrict__ v_feat,
    const float* __restrict__ in_w, const float* __restrict__ in_b,
    float* __restrict__ kh_map, float* __restrict__ vh_map) {
  const int lane = threadIdx.x & 31;
  const int gw   = blockIdx.x * 8 + (threadIdx.x >> 5);
  const int mt = gw >> 3, rem = gw & 7;
  const int kv = rem >> 2, np = rem & 3, n0 = np * 32;
  const int r0 = mt * 16;
  const float* A = (kv ? v_feat : k_feat) + (size_t)r0 * CIN;
  const float* W = in_w + (kv ? 2 : 1) * CIN * CIN;  // Wk rows 128.., Wv rows 256..
  const float* bias = in_b + (kv ? 2 : 1) * CIN;
  float* out = (kv ? vh_map : kh_map) + (size_t)r0 * CIN;

  v8f c0 = {0,0,0,0,0,0,0,0}, c1 = c0;
  gemm_loop(A, CIN, W + n0 * CIN, W + (n0 + 16) * CIN, CIN, CIN, lane, c0, c1);

  const int mr = lane & 15, mhi = (lane >> 4) << 3;
#pragma unroll
  for (int t = 0; t < 2; t++) {
    const v8f acc = t ? c1 : c0;
    const int col = n0 + t * 16 + mr;
    const float bb = bias[col];
#pragma unroll
    for (int v = 0; v < 8; v++) out[(v + mhi) * CIN + col] = acc[v] + bb;
  }
}

// ------------- Kernel 3: q = tri(gathered) @ q_w.T + q_b + pe(n) ------------
__global__ __launch_bounds__(256) void q_kernel(
    const float* __restrict__ tp, const float* __restrict__ q_w,
    const float* __restrict__ q_b, float* __restrict__ q_buf) {
  const int lane = threadIdx.x & 31;
  const int gw   = blockIdx.x * 8 + (threadIdx.x >> 5);   // [0,16384)
  const int mt = gw >> 2, np = gw & 3, n0 = np * 32;
  const int r0 = mt * 16;

  const int mr = lane & 15, kg = (lane >> 4) << 1;
  const int r = r0 + mr;
  const int b = r >> 15, n = r & 32767;
  const int x = n >> 10, y = (n >> 5) & 31, z = n & 31;
  const int tb = b * (64 * 96 * 32);
  // tri[k]      (k<64):  T[b][k   ][z   ][x]
  // tri[64+k]:           T[b][k   ][32+y][x]
  // tri[128+k]:          T[b][k   ][64+z][y]
  const int off0 = tb + z * 32 + x;
  const int off1 = tb + (32 + y) * 32 + x;
  const int off2 = tb + (64 + z) * 32 + y;
  const float* w0p = q_w + (n0 + mr) * 192 + kg;
  const float* w1p = w0p + 16 * 192;

  v8f c0 = {0,0,0,0,0,0,0,0}, c1 = c0;
#pragma unroll 4
  for (int k0 = 0; k0 < 192; k0 += 4) {
    const int kk = k0 + kg;             // even, so kk/kk+1 share a plane
    const int pl = kk >> 6;
    const int cc = kk & 63;
    const float* ap = tp + (pl == 0 ? off0 : (pl == 1 ? off1 : off2)) + cc * 3072;
    v2f a; a.x = ap[0]; a.y = ap[3072];
    v2f b0 = *(const v2f*)(w0p + k0);
    v2f b1 = *(const v2f*)(w1p + k0);
    c0 = wmma_f32(a, b0, c0);
    c1 = wmma_f32(a, b1, c1);
  }

  float* out = q_buf + (size_t)r0 * CIN;
  const int mhi = (lane >> 4) << 3;
#pragma unroll
  for (int t = 0; t < 2; t++) {
    const v8f acc = t ? c1 : c0;
    const int col = n0 + t * 16 + mr;
    const float bb = q_b[col];
    const float div = __expf(-PE_C * (float)(col >> 1));
    const bool odd = col & 1;
#pragma unroll
    for (int v = 0; v < 8; v++) {
      const int m = v + mhi;
      const int nn = (r0 + m) & 32767;
      const float ang = (float)nn * div;
      const float pe = odd ? cosf(ang) : sinf(ang);
      out[m * CIN + col] = acc[v] + bb + pe;
    }
  }
}

// --------- Kernel 4 (generic): out = A @ W.T + b, A:(65536,128), W:(128,128) -
__global__ __launch_bounds__(256) void lin128_kernel(
    const float* __restrict__ A_, const float* __restrict__ W,
    const float* __restrict__ bias, float* __restrict__ out_) {
  const int lane = threadIdx.x & 31;
  const int gw   = blockIdx.x * 8 + (threadIdx.x >> 5);   // [0,16384)
  const int mt = gw >> 2, np = gw & 3, n0 = np * 32;
  const int r0 = mt * 16;
  const float* A = A_ + (size_t)r0 * CIN;
  float* out = out_ + (size_t)r0 * CIN;

  v8f c0 = {0,0,0,0,0,0,0,0}, c1 = c0;
  gemm_loop(A, CIN, W + n0 * CIN, W + (n0 + 16) * CIN, CIN, CIN, lane, c0, c1);

  const int mr = lane & 15, mhi = (lane >> 4) << 3;
#pragma unroll
  for (int t = 0; t < 2; t++) {
    const v8f acc = t ? c1 : c0;
    const int col = n0 + t * 16 + mr;
    const float bb = bias[col];
#pragma unroll
    for (int v = 0; v < 8; v++) out[(v + mhi) * CIN + col] = acc[v] + bb;
  }
}

// ------- Kernel 5: fused project + bilinear sample + attention (J=5) --------
__global__ __launch_bounds__(256) void attn_kernel(
    const float* __restrict__ qh, const float* __restrict__ khm,
    const float* __restrict__ vhm, const float* __restrict__ proj,
    float* __restrict__ outo) {
  const int tid = blockIdx.x * 256 + threadIdx.x;
  const int h  = tid & 7;            // head
  const int vg = tid >> 3;           // [0, B*N)
  const int b = vg >> 15, n = vg & 32767;
  const int x = n >> 10, y = (n >> 5) & 31, z = n & 31;
  const float s2 = 2.0f * (1.0f + 0.1f + 1e-5f);
  const float vx = ((float)x * (1.0f / 31.0f) - 0.5f) * s2;
  const float vy = ((float)y * (1.0f / 31.0f) - 0.5f) * s2;
  const float vz = ((float)z * (1.0f / 31.0f) - 0.5f) * s2;

  const int qoff = vg * CIN + h * HD;
  v4f q[4], o[4];
#pragma unroll
  for (int g = 0; g < 4; g++) {
    q[g] = *(const v4f*)(qh + qoff + g * 4);
    o[g] = (v4f){0.f, 0.f, 0.f, 0.f};
  }
  float mrun = -3.0e38f, srun = 0.f;

  for (int j = 0; j < NIMG; j++) {
    const float* pm = proj + (b * NIMG + j) * 16;
    const float pc0 = pm[0] * vx + pm[1] * vy + pm[2]  * vz + pm[3];
    const float pc1 = pm[4] * vx + pm[5] * vy + pm[6]  * vz + pm[7];
    const float pc2 = pm[8] * vx + pm[9] * vy + pm[10] * vz + pm[11];
    const float px = pc0 / pc2, py = pc1 / pc2;
    const float gx = fminf(fmaxf((px * (1.0f / 223.0f) - 0.5f) * 2.0f, -1.f), 1.f);
    const float gy = fminf(fmaxf((py * (1.0f / 223.0f) - 0.5f) * 2.0f, -1.f), 1.f);
    const float ux = (gx + 1.f) * 0.5f * 15.f;
    const float uy = (gy + 1.f) * 0.5f * 15.f;
    const float fx0 = floorf(ux), fy0 = floorf(uy);
    int x0 = (int)fx0, y0 = (int)fy0;
    const float wx = ux - fx0, wy = uy - fy0;
    int x1 = max(min(x0 + 1, 15), 0), y1 = max(min(y0 + 1, 15), 0);
    x0 = max(min(x0, 15), 0);  y0 = max(min(y0, 15), 0);
    const float w00 = (1.f - wx) * (1.f - wy), w01 = wx * (1.f - wy);
    const float w10 = (1.f - wx) * wy,         w11 = wx * wy;
    const int mb = ((b * NIMG + j) * 256) * CIN + h * HD;
    const int i00 = mb + (y0 * 16 + x0) * CIN, i01 = mb + (y0 * 16 + x1) * CIN;
    const int i10 = mb + (y1 * 16 + x0) * CIN, i11 = mb + (y1 * 16 + x1) * CIN;

    float lg = 0.f;
    v4f ks[4];
#pragma unroll
    for (int g = 0; g < 4; g++) {
      const int go = g * 4;
      v4f t = w00 * *(const v4f*)(khm + i00 + go)
            + w01 * *(const v4f*)(khm + i01 + go)
            + w10 * *(const v4f*)(khm + i10 + go)
            + w11 * *(const v4f*)(khm + i11 + go);
      ks[g] = t;
      lg += q[g].x * t.x + q[g].y * t.y + q[g].z * t.z + q[g].w * t.w;
    }
    (void)ks;
    lg *= 0.25f;  // 1/sqrt(HD)

    const float nm = fmaxf(mrun, lg);
    const float sc = expf(mrun - nm);
    const float ww = expf(lg - nm);
    srun = srun * sc + ww;
#pragma unroll
    for (int g = 0; g < 4; g++) {
      const int go = g * 4;
      v4f tv = w00 * *(const v4f*)(vhm + i00 + go)
             + w01 * *(const v4f*)(vhm + i01 + go)
             + w10 * *(const v4f*)(vhm + i10 + go)
             + w11 * *(const v4f*)(vhm + i11 + go);
      o[g] = o[g] * sc + ww * tv;
    }
    mrun = nm;
  }
  const float inv = 1.f / srun;
#pragma unroll
  for (int g = 0; g < 4; g++) *(v4f*)(outo + qoff + g * 4) = o[g] * inv;
}

// ----------------- Kernel 6: axis means into triplanes ----------------------
__global__ __launch_bounds__(256) void mean_kernel(const float* __restrict__ vol,
                                                   float* __restrict__ planes) {
  const int t = blockIdx.x * 256 + threadIdx.x;   // [0, 786432)
  const int c  = t & 127;
  const int rq = t >> 7;
  const int qq = rq & 31;
  const int bp = rq >> 5;
  const int p  = bp % 96;
  const int b  = bp / 96;
  const int g = p >> 5, pi = p & 31;
  int base, stride;
  if (g == 0)      { base = qq * 1024 + pi;      stride = 32;   }  // xz: mean over y
  else if (g == 1) { base = qq * 1024 + pi * 32; stride = 1;    }  // xy: mean over z
  else             { base = qq * 32 + pi;        stride = 1024; }  // yz: mean over x
  const float* src = vol + ((size_t)(b * NVOX + base)) * CIN + c;
  float s = 0.f;
#pragma unroll 8
  for (int i = 0; i < 32; i++) s += src[(size_t)i * stride * CIN];
  planes[t] = s * (1.0f / 32.0f);
}

// --------- Kernel 7: out = planes @ proj_w.T + b, transposed store ----------
__global__ __launch_bounds__(256) void final_kernel(
    const float* __restrict__ planes, const float* __restrict__ proj_w,
    const float* __restrict__ proj_b, float* __restrict__ out) {
  const int lane = threadIdx.x & 31;
  const int gw   = blockIdx.x * 8 + (threadIdx.x >> 5);   // [0,768)
  const int mt = gw >> 1, np = gw & 1, n0 = np * 32;
  const int r0 = mt * 16;
  const float* A = planes + (size_t)r0 * CIN;

  v8f c0 = {0,0,0,0,0,0,0,0}, c1 = c0;
  gemm_loop(A, CIN, proj_w + n0 * CIN, proj_w + (n0 + 16) * CIN, CIN, CIN, lane, c0, c1);

  const int mr = lane & 15, mhi = (lane >> 4) << 3;
#pragma unroll
  for (int t = 0; t < 2; t++) {
    const v8f acc = t ? c1 : c0;
    const int co = n0 + t * 16 + mr;
    const float bb = proj_b[co];
#pragma unroll
    for (int v = 0; v < 8; v++) {
      const int r = r0 + v + mhi;           // (b*96+p)*32+q
      const int b = r / 3072, rem = r % 3072;
      const int p = rem >> 5, qq = rem & 31;
      out[((b * 64 + co) * 96 + p) * 32 + qq] = acc[v] + bb;
    }
  }
}

extern "C" void kernel_launch(void* const* d_in, const int* in_sizes, int n_in,
                              void* d_out, int out_size, void* d_ws, size_t ws_size,
                              hipStream_t stream) {
  (void)in_sizes; (void)n_in; (void)out_size; (void)ws_size;
  const float* triplane  = (const float*)d_in[0];
  const float* image_feat= (const float*)d_in[1];
  const float* proj_mat  = (const float*)d_in[2];
  // d_in[3] valid_frames: all-ones, unused by reference math
  const float* k_w   = (const float*)d_in[4];
  const float* k_b   = (const float*)d_in[5];
  const float* q_w   = (const float*)d_in[6];
  const float* q_b   = (const float*)d_in[7];
  const float* v_w   = (const float*)d_in[8];
  const float* v_b   = (const float*)d_in[9];
  const float* in_w  = (const float*)d_in[10];
  const float* in_b  = (const float*)d_in[11];
  const float* out_w = (const float*)d_in[12];
  const float* out_b = (const float*)d_in[13];
  const float* proj_w= (const float*)d_in[14];
  const float* proj_b= (const float*)d_in[15];

  float* ws = (float*)d_ws;
  float* k_feat = ws;                    // 327,680
  float* v_feat = ws + 327680;           // 327,680
  float* kh_map = ws + 655360;           // 327,680
  float* vh_map = ws + 983040;           // 327,680
  float* q_buf  = ws + 1310720;          // 8,388,608
  float* qh_buf = ws + 9699328;          // 8,388,608
  float* planes = ws + 18087936;         // 786,432  (total 75.5 MB)
  float* attn_buf = qh_buf;              // safe alias: each element RAW by its own thread
  float* vol_buf  = q_buf;               // safe alias: q consumed before out-proj

  kv_feat_kernel<<<160,  256, 0, stream>>>(image_feat, k_w, k_b, v_w, v_b, k_feat, v_feat);
  kv_head_kernel<<<160,  256, 0, stream>>>(k_feat, v_feat, in_w, in_b, kh_map, vh_map);
  q_kernel      <<<2048, 256, 0, stream>>>(triplane, q_w, q_b, q_buf);
  lin128_kernel <<<2048, 256, 0, stream>>>(q_buf, in_w /*Wq rows 0..127*/, in_b, qh_buf);
  attn_kernel   <<<2048, 256, 0, stream>>>(qh_buf, kh_map, vh_map, proj_mat, attn_buf);
  lin128_kernel <<<2048, 256, 0, stream>>>(attn_buf, out_w, out_b, vol_buf);
  mean_kernel   <<<3072, 256, 0, stream>>>(vol_buf, planes);
  final_kernel  <<<96,   256, 0, stream>>>(planes, proj_w, proj_b, (float*)d_out);
}